// EMASlopeFilter_64020782514583
// MI455X (gfx1250) — compile-verified
//
#include <hip/hip_runtime.h>

#define T_LEN  4096
#define NB     64
#define NF     64
#define SEG    128
#define NSEG   32
#define NEARLY 4            // segments where the adjust divide still matters (t < 512)
#define L_LAG  25
#define ALPHA  0.076923076923076927f   // 2/26
#define OM     0.92307692307692313f    // 24/26
#define NSERIES (NB * NF)              // 4096
#define NTOT ((size_t)NB * T_LEN * NF) // 16777216 elements per output

typedef __attribute__((ext_vector_type(2))) float v2f;
typedef __attribute__((ext_vector_type(8))) float v8f;

// ---------------------------------------------------------------------------
// Pass 1: per (batch, feature, segment) compute Q (segment output with zero
// carry-in) and P (product of per-step carry coefficients).
// ---------------------------------------------------------------------------
__global__ void __launch_bounds__(256)
ema_pass1(const float* __restrict__ x, float* __restrict__ Q, float* __restrict__ P) {
  int tid = blockIdx.x * blockDim.x + threadIdx.x;   // 131072 threads
  int f   = tid & 63;
  int seg = (tid >> 6) & 31;
  int b   = tid >> 11;
  int t0  = seg * SEG;
  const float* xp = x + ((size_t)b * T_LEN + t0) * NF + f;

  float q, p;
  if (seg >= NEARLY) {
    // steady state: w == 1.0f exactly, pure EMA with constant coefficients
    q = 0.f;
    #pragma unroll 4
    for (int i = 0; i < SEG; ++i) {
      __builtin_prefetch(xp + (size_t)(i + 16) * NF, 0, 0);  // global_prefetch_b8
      q = fmaf(OM, q, ALPHA * xp[(size_t)i * NF]);
    }
    p = __powf(OM, (float)SEG);
  } else if (seg == 0) {
    // e_0 = x_0, then adjusted recurrence
    q = xp[0];
    float omp = OM;                                   // om^(t+1) tracker
    for (int i = 1; i < SEG; ++i) {
      omp *= OM;
      float w = fmaxf(1.f - omp, 1e-10f);
      q = (ALPHA * xp[(size_t)i * NF] + OM * q) / w;
    }
    p = 0.f;                                          // no carry-in for seg 0
  } else {
    q = 0.f; p = 1.f;
    float omp = __powf(OM, (float)t0);
    for (int i = 0; i < SEG; ++i) {
      omp *= OM;
      float w  = fmaxf(1.f - omp, 1e-10f);
      float rw = 1.f / w;
      q = (ALPHA * xp[(size_t)i * NF] + OM * q) * rw;
      p *= OM * rw;
    }
  }
  int sidx = seg * NSERIES + b * NF + f;
  Q[sidx] = q;
  P[sidx] = p;
}

// ---------------------------------------------------------------------------
// Pass 2: per-series scan over the 32 segment (P,Q) pairs -> carry-in per seg
// ---------------------------------------------------------------------------
__global__ void __launch_bounds__(256)
ema_pass2(const float* __restrict__ Q, const float* __restrict__ P,
          float* __restrict__ Ein) {
  int s = blockIdx.x * blockDim.x + threadIdx.x;      // 4096 series
  float c = 0.f;
  #pragma unroll
  for (int k = 0; k < NSEG; ++k) {
    int idx = k * NSERIES + s;
    Ein[idx] = c;
    c = fmaf(P[idx], c, Q[idx]);
  }
}

// ---------------------------------------------------------------------------
// Pass 3a: expand early segments (t < 512) with the adjust divide, scalar.
// ---------------------------------------------------------------------------
__global__ void __launch_bounds__(256)
ema_pass3_early(const float* __restrict__ x, const float* __restrict__ Ein,
                float* __restrict__ oema) {
  int tid = blockIdx.x * blockDim.x + threadIdx.x;    // 16384 threads
  int f   = tid & 63;
  int seg = (tid >> 6) & 3;
  int b   = tid >> 8;
  int t0  = seg * SEG;
  const float* xp = x    + ((size_t)b * T_LEN + t0) * NF + f;
  float*       op = oema + ((size_t)b * T_LEN + t0) * NF + f;

  float e, omp;
  int istart;
  if (seg == 0) {
    e = xp[0];
    op[0] = e;
    omp = OM;
    istart = 1;
  } else {
    e = Ein[seg * NSERIES + b * NF + f];
    omp = __powf(OM, (float)t0);
    istart = 0;
  }
  for (int i = istart; i < SEG; ++i) {
    omp *= OM;
    float w = fmaxf(1.f - omp, 1e-10f);
    e = (ALPHA * xp[(size_t)i * NF] + OM * e) / w;
    op[(size_t)i * NF] = e;
  }
}

// ---------------------------------------------------------------------------
// Pass 3b: steady-state segments via WMMA. Each wave handles 16 features of
// one segment. A 16-step chunk is E = C16x16 * X + (om^{i+1} * e_in), done
// as 4 chained V_WMMA_F32_16X16X4_F32 (K-slices) accumulating onto the carry.
// Chunk loop fully unrolled so every global access is base + immediate.
// ---------------------------------------------------------------------------
__global__ void __launch_bounds__(128)
ema_pass3_wmma(const float* __restrict__ x, const float* __restrict__ Ein,
               float* __restrict__ oema) {
  int wave = blockIdx.x * (blockDim.x >> 5) + (threadIdx.x >> 5); // 7168 waves
  int lane = threadIdx.x & 31;
  int n    = lane & 15;        // column (feature within group) / A row M
  int half = lane >> 4;        // 0: lanes 0-15, 1: lanes 16-31

  const int NSTEADY = NSEG - NEARLY;               // 28
  int segr = wave % NSTEADY;
  int fg   = (wave / NSTEADY) & 3;
  int b    = wave / (NSTEADY * 4);
  int seg  = segr + NEARLY;
  int f    = fg * 16 + n;
  int tseg = seg * SEG;

  size_t bbase = (size_t)b * T_LEN * NF;
  // All per-chunk offsets below are compile-time immediates from these bases:
  // B loads:  xbase + (ci*16 + 4k)*NF  (+NF for the second K row)
  // D stores: obase + (ci*16 + r)*NF
  const float* xbase = x    + bbase + ((size_t)(tseg + 2 * half)) * NF + f;
  float*       obase = oema + bbase + ((size_t)(tseg + 8 * half)) * NF + f;

  // om^0 .. om^16 (constant-foldable)
  float omp[17];
  omp[0] = 1.f;
  #pragma unroll
  for (int i = 1; i < 17; ++i) omp[i] = omp[i - 1] * OM;
  float cf = half ? omp[8] : 1.f;                  // om^{8*half}

  // A-matrix K-slices, ISA layout: lane M = n, K = 4k + 2*half + {0,1}
  // coeff(i,j) = (j<=i) ? alpha*om^(i-j) : 0
  v2f a[4];
  #pragma unroll
  for (int k = 0; k < 4; ++k) {
    int j0 = 4 * k + 2 * half;
    a[k].x = (j0     <= n) ? ALPHA * __powf(OM, (float)(n - j0))     : 0.f;
    a[k].y = (j0 + 1 <= n) ? ALPHA * __powf(OM, (float)(n - j0 - 1)) : 0.f;
  }

  float ein = Ein[seg * NSERIES + b * NF + f];     // carry into segment

  #pragma unroll
  for (int ci = 0; ci < 8; ++ci) {                 // 8 chunks of 16 steps
    // C[i][n] = om^{i+1} * e_in[n]; D-layout row M = r + 8*half
    v8f acc;
    #pragma unroll
    for (int r = 0; r < 8; ++r) acc[r] = ein * (cf * omp[r + 1]);

    #pragma unroll
    for (int k = 0; k < 4; ++k) {
      v2f bv;
      bv.x = xbase[(ci * 16 + 4 * k) * NF];        // immediate offsets
      bv.y = xbase[(ci * 16 + 4 * k + 1) * NF];
      acc = __builtin_amdgcn_wmma_f32_16x16x4_f32(
          false, a[k], false, bv, (short)0, acc, false, false);
    }

    // scatter D back: lane(l, vgpr r) -> M = r + 8*half, N = n
    #pragma unroll
    for (int r = 0; r < 8; ++r)
      obase[(ci * 16 + r) * NF] = acc[r];          // immediate offsets

    // chunk-end carry: row M=15 lives in VGPR7 on lanes 16..31
    float eend = acc[7];
    ein = __shfl(eend, 16 + n, 32);
  }
}

// ---------------------------------------------------------------------------
// Pass 4: slope + signals, float4-vectorized streaming pass (ema is L2-hot).
// ---------------------------------------------------------------------------
__global__ void __launch_bounds__(256)
ema_pass4(const float* __restrict__ oema, float* __restrict__ out) {
  int tid = blockIdx.x * blockDim.x + threadIdx.x;  // NTOT/4 threads
  const float4* e4 = reinterpret_cast<const float4*>(oema);
  float4 ev = e4[tid];
  int t = (tid >> 4) & (T_LEN - 1);                 // 4 floats share one t
  float4 lg = make_float4(0.f, 0.f, 0.f, 0.f);
  if (t >= L_LAG) lg = e4[tid - (L_LAG * NF / 4)];

  float4 sl = make_float4(ev.x - lg.x, ev.y - lg.y, ev.z - lg.z, ev.w - lg.w);
  float4 ab, bl, bt;
  ab.x = (sl.x >  15.0f) ? 1.f : 0.f;  ab.y = (sl.y >  15.0f) ? 1.f : 0.f;
  ab.z = (sl.z >  15.0f) ? 1.f : 0.f;  ab.w = (sl.w >  15.0f) ? 1.f : 0.f;
  bl.x = (sl.x < -15.0f) ? 1.f : 0.f;  bl.y = (sl.y < -15.0f) ? 1.f : 0.f;
  bl.z = (sl.z < -15.0f) ? 1.f : 0.f;  bl.w = (sl.w < -15.0f) ? 1.f : 0.f;
  bt.x = (sl.x >= -15.0f && sl.x <= 15.0f) ? 1.f : 0.f;
  bt.y = (sl.y >= -15.0f && sl.y <= 15.0f) ? 1.f : 0.f;
  bt.z = (sl.z >= -15.0f && sl.z <= 15.0f) ? 1.f : 0.f;
  bt.w = (sl.w >= -15.0f && sl.w <= 15.0f) ? 1.f : 0.f;

  const int NQ = (int)(NTOT / 4);
  float4* o4 = reinterpret_cast<float4*>(out);
  o4[NQ * 1 + tid] = sl;
  o4[NQ * 2 + tid] = ab;
  o4[NQ * 3 + tid] = bl;
  o4[NQ * 4 + tid] = bt;
}

// ---------------------------------------------------------------------------
extern "C" void kernel_launch(void* const* d_in, const int* in_sizes, int n_in,
                              void* d_out, int out_size, void* d_ws, size_t ws_size,
                              hipStream_t stream) {
  (void)in_sizes; (void)n_in; (void)out_size; (void)d_ws; (void)ws_size;
  const float* x  = (const float*)d_in[0];
  float* out  = (float*)d_out;
  float* oema = out;                                  // output section 0 (ema)

  // Scratch lives in the slope section of d_out; pass 4 fully overwrites it.
  float* scratch = out + NTOT;
  float* Q   = scratch;
  float* P   = scratch + 1 * NSEG * NSERIES;
  float* Ein = scratch + 2 * NSEG * NSERIES;          // 1.5 MB total

  ema_pass1      <<<  512, 256, 0, stream>>>(x, Q, P);          // 131072 thr
  ema_pass2      <<<   16, 256, 0, stream>>>(Q, P, Ein);        //   4096 thr
  ema_pass3_early<<<   64, 256, 0, stream>>>(x, Ein, oema);     //  16384 thr
  ema_pass3_wmma <<< 1792, 128, 0, stream>>>(x, Ein, oema);     //   7168 waves
  ema_pass4      <<<16384, 256, 0, stream>>>(oema, out);        //    4 M thr
}